// BERTEmbedding_64226940944419
// MI455X (gfx1250) — compile-verified
//
#include <hip/hip_runtime.h>
#include <math.h>

// ---------------------------------------------------------------------------
// BERT embedding for MI455X (gfx1250, wave32).
//   out[:, 0:256]   = input[B,S,10] @ W^T[10,256] + b   (fp32 WMMA 16x16x4)
//   out[:, 256:512] = sinusoidal PE row (doy==0 -> zeros), computed in-kernel
// Store-bandwidth bound (256 MB out @ 23.3 TB/s ~ 11 us):
//   * branchless inner loop (B-fragments from zero-padded W staged in LDS)
//   * WMMA D tiles bounced through LDS -> wide b128 nontemporal stores
// ---------------------------------------------------------------------------

typedef __attribute__((ext_vector_type(2))) float v2f;
typedef __attribute__((ext_vector_type(4))) float v4f;
typedef __attribute__((ext_vector_type(8))) float v8f;

#define NUM_FEATURES 10
#define KPAD         12               // features padded to 3 K-chunks of 4
#define EMBED        256
#define OUTDIM       512
#define NTOK         (256 * 512)      // B*S
#define WAVES_PER_BLOCK 8
#define TOKENS_PER_WAVE 16
#define TOKENS_PER_BLOCK (WAVES_PER_BLOCK * TOKENS_PER_WAVE)  // 128

__global__ __launch_bounds__(256) void bert_embed_kernel(
    const float* __restrict__ in,    // [NTOK, 10]
    const int*   __restrict__ doy,   // [NTOK]
    const float* __restrict__ W,     // [256, 10]  (row = embed dim, col = feature)
    const float* __restrict__ bias,  // [256]
    float*       __restrict__ out)   // [NTOK, 512]
{
    __shared__ __align__(16) float sWp[EMBED * KPAD];          // 12 KB, zero-padded W
    __shared__ __align__(16) float sBias[EMBED];               //  1 KB
    __shared__ __align__(16) float sD[WAVES_PER_BLOCK][16*64]; // 32 KB, D staging

    const int lane = threadIdx.x & 31;
    const int wave = threadIdx.x >> 5;
    const int t0   = (blockIdx.x * WAVES_PER_BLOCK + wave) * TOKENS_PER_WAVE;

    const int m  = lane & 15;         // A row / B col / D col within 16
    const int kb = (lane >> 4) * 2;   // K sub-offset inside a 4-wide K chunk: 0 or 2

    // ---- Cooperative stage: zero-padded W^T and bias into LDS (once/block) ----
    {
        const int e = threadIdx.x;                 // 0..255 = embed dim
        const float* wr = W + (size_t)e * NUM_FEATURES;
        float* wp = sWp + e * KPAD;
#pragma unroll
        for (int i = 0; i < NUM_FEATURES / 2; ++i) {
            v2f w = *(const v2f*)(wr + 2 * i);     // e*10 even -> 8B aligned
            wp[2 * i]     = w.x;
            wp[2 * i + 1] = w.y;
        }
        wp[10] = 0.0f;                             // K padding
        wp[11] = 0.0f;
        sBias[e] = bias[e];
    }
    __syncthreads();

    // ---- A fragments: 16 tokens x 12 padded features, 3 K-chunks of 4 --------
    // V_WMMA_F32_16X16X4_F32 A layout: lanes 0-15 -> {K0,K1}, lanes 16-31 -> {K2,K3}
    const float* arow = in + (size_t)(t0 + m) * NUM_FEATURES;
    v2f a0 = *(const v2f*)(arow + kb);             // k = {0,1} | {2,3}
    v2f a1 = *(const v2f*)(arow + 4 + kb);         // k = {4,5} | {6,7}
    v2f a2;
    if (kb == 0) {
        a2 = *(const v2f*)(arow + 8);              // k = {8,9}
    } else {
        a2.x = 0.0f; a2.y = 0.0f;                  // k = {10,11} : zero padding
    }

    const int row_off = (lane >> 4) * 8;           // D: lanes 16-31 hold rows M+8
    float* sd = &sD[wave][0];

    // ---- Linear part: 16 N-tiles in 4 groups of 4; drain each group as b128 --
    for (int g = 0; g < 4; ++g) {
#pragma unroll
        for (int tt = 0; tt < 4; ++tt) {
            const int tile = g * 4 + tt;
            const int n    = tile * 16 + m;
            const float* wp = sWp + n * KPAD + kb; // even offsets -> ds_load_b64
            v2f b0 = *(const v2f*)(wp);
            v2f b1 = *(const v2f*)(wp + 4);
            v2f b2 = *(const v2f*)(wp + 8);        // zeros pre-padded, no branch

            const float bv = sBias[n];             // bias fused via C init
            v8f c = { bv, bv, bv, bv, bv, bv, bv, bv };

            c = __builtin_amdgcn_wmma_f32_16x16x4_f32(false, a0, false, b0,
                                                      (short)0, c, false, false);
            c = __builtin_amdgcn_wmma_f32_16x16x4_f32(false, a1, false, b1,
                                                      (short)0, c, false, false);
            c = __builtin_amdgcn_wmma_f32_16x16x4_f32(false, a2, false, b2,
                                                      (short)0, c, false, false);

            // Stage D tile into [16 tokens][64 cols] wave-private LDS patch.
#pragma unroll
            for (int v = 0; v < 8; ++v) {
                sd[(v + row_off) * 64 + tt * 16 + m] = c[v];
            }
        }
        // Drain 16x64 group: each b128 covers 4 consecutive embed dims; lanes
        // 0-15 / 16-31 form two 256B contiguous row segments per instruction.
        // Per-wave LDS ops are in-order -> RAW through sd is safe w/o barrier.
#pragma unroll
        for (int i = 0; i < 8; ++i) {
            const int idx  = lane * 4 + i * 128;   // flat index into 16x64
            const int tokr = idx >> 6;             // 0..15
            const int col  = idx & 63;
            v4f d = *(const v4f*)(sd + idx);       // ds_load_b128, 16B aligned
            __builtin_nontemporal_store(
                d, (v4f*)(out + (size_t)(t0 + tokr) * OUTDIM + g * 64 + col));
        }
    }

    // ---- Positional-embedding part -------------------------------------------
    // pe[d][e]: d==0 -> 0; else pos=d-1, i=e>>1,
    //   freq_i = exp(-(2i)*ln(10000)/256), e even -> sin(pos*freq), odd -> cos.
    const float kFreq = 0.035977892944f;           // ln(10000)/256
    const int e0a = lane * 4;                      // e in [0,128)
    const int e0b = 128 + lane * 4;                // e in [128,256)
    const float f0a = __expf(-kFreq * (float)(e0a));
    const float f1a = __expf(-kFreq * (float)(e0a + 2));
    const float f0b = __expf(-kFreq * (float)(e0b));
    const float f1b = __expf(-kFreq * (float)(e0b + 2));

#pragma unroll 4
    for (int mm = 0; mm < TOKENS_PER_WAVE; ++mm) {
        const int tok = t0 + mm;
        const int d   = doy[tok];                  // wave-uniform -> broadcast
        const float pos  = (float)(d - 1);
        const float mask = (d == 0) ? 0.0f : 1.0f;
        float* pbase = out + (size_t)tok * OUTDIM + EMBED;

        float s0, c0, s1, c1;
        v4f v;

        __sincosf(pos * f0a, &s0, &c0);
        __sincosf(pos * f1a, &s1, &c1);
        v.x = s0 * mask; v.y = c0 * mask; v.z = s1 * mask; v.w = c1 * mask;
        __builtin_nontemporal_store(v, (v4f*)(pbase + e0a));

        __sincosf(pos * f0b, &s0, &c0);
        __sincosf(pos * f1b, &s1, &c1);
        v.x = s0 * mask; v.y = c0 * mask; v.z = s1 * mask; v.w = c1 * mask;
        __builtin_nontemporal_store(v, (v4f*)(pbase + e0b));
    }
}

extern "C" void kernel_launch(void* const* d_in, const int* in_sizes, int n_in,
                              void* d_out, int out_size, void* d_ws, size_t ws_size,
                              hipStream_t stream) {
    (void)in_sizes; (void)n_in; (void)out_size; (void)d_ws; (void)ws_size;
    const float* in  = (const float*)d_in[0];   // [256,512,10] fp32
    const int*   doy = (const int*)  d_in[1];   // [256,512]    int32
    const float* W   = (const float*)d_in[2];   // [256,10]     fp32
    const float* b   = (const float*)d_in[3];   // [256]        fp32
    float* out = (float*)d_out;                 // [256,512,512] fp32

    dim3 grid(NTOK / TOKENS_PER_BLOCK);         // 1024 blocks, exact cover
    dim3 block(256);                            // 8 waves of 32
    hipLaunchKernelGGL(bert_embed_kernel, grid, block, 0, stream,
                       in, doy, W, b, out);
}